// InfoNce_3461743640727
// MI455X (gfx1250) — compile-verified
//
#include <hip/hip_runtime.h>

// CDNA5 / gfx1250 InfoNCE loss.
// One wave32 per group: G = F*F^T via V_WMMA_F32_16X16X4_F32 (f32-exact),
// norms from diag(G), per-row InfoNCE epilogue, deterministic reduction.
// Bandwidth-bound: 134 MB single pass -> ~5.8 us floor at 23.3 TB/s.

typedef __attribute__((ext_vector_type(2)))  float  v2f;
typedef __attribute__((ext_vector_type(8)))  float  v8f;

constexpr int   Cdim     = 2048;   // feature length
constexpr int   Tdim     = 4;      // trailing dim t
constexpr int   NIc      = 4;      // num_instance
constexpr int   Nrow     = 16;     // n = t*ni rows per group
constexpr int   KOUT     = 64;     // k-columns per outer iteration (1 prefetch)
constexpr int   PFDIST   = 256;    // prefetch distance in k-columns (4 KB)
constexpr float TEMP_INV = 1.0f / 0.07f;

__global__ __launch_bounds__(32)
void infonce_gram_kernel(const float* __restrict__ feat,
                         float* __restrict__ part, int g_total)
{
    const int g = blockIdx.x;
    if (g >= g_total) return;                 // uniform branch; EXEC stays all-1s

    const int lane = threadIdx.x;             // wave32
    const int r    = lane & 15;               // row index (M for A, N for B)
    const int h    = lane >> 4;               // lane half -> K pairing
    const int nn   = r & 3;                   // instance within group
    const int tt   = r >> 2;                  // t index
    // f[g][r][k] == features[g*NI + nn][k][tt]; element k stride = Tdim floats
    const float* __restrict__ rowp =
        feat + ((size_t)(g * NIc + nn) * (size_t)(Cdim * Tdim)) + tt;

    v8f acc = {0.f, 0.f, 0.f, 0.f, 0.f, 0.f, 0.f, 0.f};

    // f32 A 16x4 layout: VGPR j, lane-half h holds K = 2*h + j.
    // B (4x16) mirrors A with lane=N, so for the symmetric Gram A==B per lane.
    const int koff = 2 * h;

    // Outer loop: one L2 prefetch per 64 k-columns; inner loop is pure
    // load+WMMA (no conditionals -> no per-chunk SALU/branch overhead).
    for (int kb0 = 0; kb0 < Cdim; kb0 += KOUT) {
        if (kb0 + PFDIST < Cdim)
            __builtin_prefetch((const void*)(rowp + (size_t)(kb0 + PFDIST) * Tdim), 0, 0);
        #pragma unroll
        for (int kq = 0; kq < KOUT; kq += 4) {
            const int k = kb0 + kq + koff;
            v2f ab;
            ab.x = rowp[(size_t)k * Tdim];
            ab.y = rowp[(size_t)(k + 1) * Tdim];
            acc = __builtin_amdgcn_wmma_f32_16x16x4_f32(
                /*neg_a=*/false, ab, /*neg_b=*/false, ab,
                /*c_mod=*/(short)0, acc, /*reuse_a=*/false, /*reuse_b=*/false);
        }
    }

    // ---- epilogue: G (16x16) -> LDS ----
    __shared__ float gm[Nrow * Nrow];
    __shared__ float rsn[Nrow];
    __shared__ float rowloss[Nrow];

    // C/D layout: VGPR v: lanes0-15 -> M=v, lanes16-31 -> M=v+8; N = lane&15.
    #pragma unroll
    for (int v = 0; v < 8; ++v)
        gm[(v + 8 * h) * Nrow + r] = acc[v];
    __syncthreads();

    if (lane < Nrow)
        rsn[lane] = 1.0f / fmaxf(sqrtf(gm[lane * Nrow + lane]), 1e-12f);
    __syncthreads();

    if (lane < Nrow) {
        const int i  = lane;
        const float ri = rsn[i];
        float lg[Nrow];
        float eneg = 0.0f;
        #pragma unroll
        for (int j = 0; j < Nrow; ++j) {
            const float cosv = gm[i * Nrow + j] * ri * rsn[j];
            lg[j] = cosv * TEMP_INV;
            if ((i >> 2) != (j >> 2))          // different label -> negative
                eneg += expf(lg[j]);
        }
        const int base = (i >> 2) << 2;        // positives: same label, j != i
        float lsum = 0.0f;
        #pragma unroll
        for (int j2 = 0; j2 < 4; ++j2) {
            const int j = base + j2;
            if (j != i)
                lsum += logf(eneg + expf(lg[j])) - lg[j];   // = -log_prob
        }
        rowloss[i] = lsum;
    }
    __syncthreads();

    if (lane == 0) {
        float s = 0.0f;
        #pragma unroll
        for (int i = 0; i < Nrow; ++i) s += rowloss[i];
        part[g] = s;
    }
}

__global__ __launch_bounds__(256)
void infonce_reduce_kernel(const float* __restrict__ part,
                           float* __restrict__ out, int g_total)
{
    __shared__ float s[256];
    const int tid = threadIdx.x;
    float v = 0.0f;
    for (int i = tid; i < g_total; i += 256) v += part[i];
    s[tid] = v;
    __syncthreads();
    #pragma unroll
    for (int off = 128; off > 0; off >>= 1) {
        if (tid < off) s[tid] += s[tid + off];
        __syncthreads();
    }
    if (tid == 0)
        out[0] = -s[0] / 48.0f;   // sum(mask_pos) = 16 rows * 3 positives
}

extern "C" void kernel_launch(void* const* d_in, const int* in_sizes, int n_in,
                              void* d_out, int out_size, void* d_ws, size_t ws_size,
                              hipStream_t stream)
{
    const float* feat = (const float*)d_in[0];
    float*       part = (float*)d_ws;                      // g_total floats of scratch
    const int g_total = in_sizes[0] / (Cdim * Tdim * NIc); // 1024 for (4096,2048,4)

    infonce_gram_kernel<<<dim3(g_total), dim3(32), 0, stream>>>(feat, part, g_total);
    infonce_reduce_kernel<<<dim3(1), dim3(256), 0, stream>>>(part, (float*)d_out, g_total);
}